// SoftProjection_1400159339082
// MI455X (gfx1250) — compile-verified
//
#include <hip/hip_runtime.h>
#include <math.h>

typedef __attribute__((ext_vector_type(2))) float v2f;
typedef __attribute__((ext_vector_type(8))) float v8f;

#define B_   4
#define N_   8192
#define M_   4096
#define F_   64
#define K_   16
#define WAVES 8
#define QPW  16               // queries per wave (one WMMA N-tile)
#define QPB  (WAVES * QPW)    // 128 queries per block

// Branchless carry-down insert into sorted-ascending top-16 held in registers.
// Pure compare+select (no fmin/fmax -> no canonicalize ops): 1 v_cmp + 4
// dual-issuable v_cndmask per stage. Caller guards with d < kd[15].
__device__ __forceinline__ void topk_insert(float (&kd)[K_], int (&ki)[K_],
                                            float d, int i) {
  float cd = d; int ci = i;
  #pragma unroll
  for (int j = 0; j < K_; ++j) {
    bool  sw  = cd < kd[j];
    float lo  = sw ? cd    : kd[j];
    float hi  = sw ? kd[j] : cd;
    int   loi = sw ? ci    : ki[j];
    int   hii = sw ? ki[j] : ci;
    kd[j] = lo; ki[j] = loi;
    cd    = hi; ci    = hii;
  }
}

__global__ __launch_bounds__(256)
void softproj_kernel(const float* __restrict__ pc,    // (B,3,N)
                     const float* __restrict__ qc,    // (B,3,M)
                     const float* __restrict__ pf,    // (B,F,N)
                     const float* __restrict__ temp,  // scalar
                     float* __restrict__ out)         // proj(B,3,M) ++ feat(B,F,M)
{
  __shared__ float s_pk[N_ * 4];             // {x,y,z,p^2} per point (128 KB)
  __shared__ float s_iw[WAVES][QPW][K_][2];  // {idx(bitcast), weight} pairs (16 KB)

  const int tid  = threadIdx.x;
  const int lane = tid & 31;
  const int wave = tid >> 5;
  const int blocks_per_batch = M_ / QPB;              // 32
  const int b     = blockIdx.x / blocks_per_batch;
  const int qbase = (blockIdx.x % blocks_per_batch) * QPB + wave * QPW;

  const float* pcb = pc + (size_t)b * 3 * N_;
  const float* qcb = qc + (size_t)b * 3 * M_;
  const float* pfb = pf + (size_t)b * F_ * N_;

  // ---- Stage per-batch point cloud as {x,y,z,p^2} quads into LDS ----
  for (int i = tid; i < N_; i += 256) {
    float x = pcb[i], y = pcb[N_ + i], z = pcb[2 * N_ + i];
    float4 v; v.x = x; v.y = y; v.z = z; v.w = x * x + y * y + z * z;
    *(float4*)&s_pk[4 * i] = v;              // ds_store_b128
  }
  __syncthreads();

  // ---- Loop-invariant B operand: columns {-2qx,-2qy,-2qz,1} so that
  //      D = A*B = p^2 - 2 q.p directly (q^2 shift cancels in top-k & softmax).
  //      Lanes 0-15 hold K rows 0,1; lanes 16-31 hold K rows 2,3.
  const int lp    = lane & 15;
  const int half  = lane >> 4;
  const int half8 = half << 3;
  const int q     = qbase + lp;
  const int aoff  = (lp << 2) + (half << 1);          // lane's A-pair offset

  v2f bq;
  if (half == 0) { bq.x = -2.0f * qcb[q];          bq.y = -2.0f * qcb[M_ + q]; }
  else           { bq.x = -2.0f * qcb[2 * M_ + q]; bq.y = 1.0f; }

  float kd[K_]; int ki[K_];
  #pragma unroll
  for (int j = 0; j < K_; ++j) { kd[j] = 3.4e38f; ki[j] = 0; }

  // ---- Sweep point tiles: one ds_load_b64 + one WMMA per 16x16 distance tile.
  //      Software-pipelined A load; tile-level min prune before any insert.
  v2f a_cur = *(const v2f*)&s_pk[aoff];
  #pragma unroll 2
  for (int t = 0; t < N_; t += 16) {
    int tn = (t + 16) & (N_ - 1);                       // wrap: avoids tail branch
    v2f a_nxt = *(const v2f*)&s_pk[(tn << 2) + aoff];

    v8f acc = {0.0f, 0.0f, 0.0f, 0.0f, 0.0f, 0.0f, 0.0f, 0.0f};
    acc = __builtin_amdgcn_wmma_f32_16x16x4_f32(
        /*neg_a=*/false, a_cur, /*neg_b=*/false, bq,
        /*c_mod=*/(short)0, acc, /*reuse_a=*/false, /*reuse_b=*/false);

    // Tile-level prune: min of the 8 candidate rows vs current worst.
    float m0 = acc[0] < acc[1] ? acc[0] : acc[1];
    float m1 = acc[2] < acc[3] ? acc[2] : acc[3];
    float m2 = acc[4] < acc[5] ? acc[4] : acc[5];
    float m3 = acc[6] < acc[7] ? acc[6] : acc[7];
    float m4 = m0 < m1 ? m0 : m1;
    float m5 = m2 < m3 ? m2 : m3;
    float tmin = m4 < m5 ? m4 : m5;

    if (tmin < kd[K_ - 1]) {
      #pragma unroll
      for (int j = 0; j < 8; ++j)
        if (acc[j] < kd[K_ - 1])
          topk_insert(kd, ki, acc[j], t + half8 + j);
    }
    a_cur = a_nxt;
  }

  // ---- Merge the two half-wave candidate lists for the same query ----
  float od[K_]; int oi[K_];
  #pragma unroll
  for (int j = 0; j < K_; ++j) {
    od[j] = __shfl_xor(kd[j], 16);
    oi[j] = __shfl_xor(ki[j], 16);
  }
  #pragma unroll
  for (int j = 0; j < K_; ++j)
    if (od[j] < kd[K_ - 1]) topk_insert(kd, ki, od[j], oi[j]);

  // ---- Exact softmax weights (delta form, matches reference) + projection ----
  float sigma = temp[0] * temp[0];
  sigma = fmaxf(sigma, 1.0e-4f);
  float invsig = 1.0f / sigma;
  float qx = qcb[q], qy = qcb[M_ + q], qz = qcb[2 * M_ + q];

  float gx[K_], gy[K_], gz[K_], de[K_];
  float dmin = 3.4e38f;
  #pragma unroll
  for (int j = 0; j < K_; ++j) {
    float4 g = *(const float4*)&s_pk[ki[j] << 2];   // ds_load_b128 gather
    gx[j] = g.x; gy[j] = g.y; gz[j] = g.z;
    float dx = g.x - qx, dy = g.y - qy, dz = g.z - qz;
    de[j] = (dx * dx + dy * dy + dz * dz) * invsig;
    dmin = dmin < de[j] ? dmin : de[j];
  }
  float w[K_], wsum = 0.0f;
  #pragma unroll
  for (int j = 0; j < K_; ++j) { w[j] = __expf(dmin - de[j]); wsum += w[j]; }
  float inv = 1.0f / wsum;
  float sx = 0.0f, sy = 0.0f, sz = 0.0f;
  #pragma unroll
  for (int j = 0; j < K_; ++j) {
    w[j] *= inv;
    sx += gx[j] * w[j]; sy += gy[j] * w[j]; sz += gz[j] * w[j];
  }

  if (half == 0) {
    out[(size_t)b * 3 * M_ +          q] = sx;
    out[(size_t)b * 3 * M_ +     M_ + q] = sy;
    out[(size_t)b * 3 * M_ + 2 * M_ + q] = sz;
    #pragma unroll
    for (int j = 0; j < K_; ++j) {
      s_iw[wave][lp][j][0] = __int_as_float(ki[j]);
      s_iw[wave][lp][j][1] = w[j];
    }
  }
  __syncthreads();

  // ---- Feature propagation: 32 lanes cover F=64 (2 features per lane) ----
  float* outF = out + (size_t)B_ * 3 * M_;
  const int f0 = lane, f1 = lane + 32;
  for (int qq = 0; qq < QPW; ++qq) {
    int qg = qbase + qq;
    float a0 = 0.0f, a1 = 0.0f;
    #pragma unroll
    for (int j = 0; j < K_; ++j) {
      float2 iw = *(const float2*)&s_iw[wave][qq][j][0];   // ds_load_b64
      int   n  = __float_as_int(iw.x);
      float wj = iw.y;
      a0 += pfb[(size_t)f0 * N_ + n] * wj;
      a1 += pfb[(size_t)f1 * N_ + n] * wj;
    }
    outF[(size_t)b * F_ * M_ + (size_t)f0 * M_ + qg] = a0;
    outF[(size_t)b * F_ * M_ + (size_t)f1 * M_ + qg] = a1;
  }
}

extern "C" void kernel_launch(void* const* d_in, const int* in_sizes, int n_in,
                              void* d_out, int out_size, void* d_ws, size_t ws_size,
                              hipStream_t stream) {
  const float* pc   = (const float*)d_in[0];
  const float* qc   = (const float*)d_in[1];
  const float* pf   = (const float*)d_in[2];
  const float* temp = (const float*)d_in[3];
  float* out = (float*)d_out;

  dim3 grid(B_ * (M_ / QPB));   // 4 * 32 = 128 blocks
  dim3 block(256);              // 8 waves, one 16-query WMMA tile per wave
  softproj_kernel<<<grid, block, 0, stream>>>(pc, qc, pf, temp, out);
}